// Net_2946347565966
// MI455X (gfx1250) — compile-verified
//
#include <hip/hip_runtime.h>
#include <hip/hip_bf16.h>

// ---------------------------------------------------------------------------
// Types for WMMA fragments (wave32, gfx1250)
// ---------------------------------------------------------------------------
typedef __attribute__((ext_vector_type(2))) float v2f;  // A/B frag of 16x16x4 f32
typedef __attribute__((ext_vector_type(8))) float v8f;  // C/D frag (16x16 f32)

// ---------------------------------------------------------------------------
// Fused locally-connected conv chain.
// Each lane owns one (b, p) column; the whole 4-stage chain (conv K=5 pad=2,
// ReLU, maxpool win=10 stride=3) runs out of LDS ping-pong buffers.
//   stage1: Cin=1, Tin=365 -> Tpool=119
//   stage2: Cin=3, Tin=119 -> Tpool=37
//   stage3: Cin=3, Tin=37  -> Tpool=10
//   stage4: Cin=3, Tin=10  -> Tpool=1
// Final 3 values per (b,p) go to c[b*1297 + co*399 + p] (the concat buffer).
// ---------------------------------------------------------------------------
#define TS 32           // LDS column stride (one float column per lane)
#define NPB 13          // ceil(399/32) position blocks

template<int CIN, int TIN, int TPOOL>
__device__ inline void lc_stage(const float* __restrict__ Wp,  // [3][CIN][5] for this p
                                const float* __restrict__ bp,  // [3]
                                const float* inb,              // LDS, lane-offset, stride TS
                                float*       outb) {           // LDS, lane-offset, stride TS
    float w[3 * CIN * 5];
#pragma unroll
    for (int i = 0; i < 3 * CIN * 5; ++i) w[i] = Wp[i];
    float bb[3];
#pragma unroll
    for (int co = 0; co < 3; ++co) bb[co] = bp[co];

#pragma unroll 1
    for (int tp = 0; tp < TPOOL; ++tp) {
        // Input window needed for pooled output tp: ti in [3tp-2, 3tp+11] (14 samples/ci)
        float xv[CIN][14];
#pragma unroll
        for (int ci = 0; ci < CIN; ++ci) {
#pragma unroll
            for (int j = 0; j < 14; ++j) {
                int ti = 3 * tp - 2 + j;
                xv[ci][j] = (ti >= 0 && ti < TIN) ? inb[(ci * TIN + ti) * TS] : 0.0f;
            }
        }
#pragma unroll
        for (int co = 0; co < 3; ++co) {
            float m = 0.0f;  // maxpool of ReLU'd values: identity is 0
#pragma unroll
            for (int dt = 0; dt < 10; ++dt) {
                float acc = bb[co];
#pragma unroll
                for (int ci = 0; ci < CIN; ++ci) {
#pragma unroll
                    for (int k = 0; k < 5; ++k)
                        acc = fmaf(xv[ci][dt + k], w[(co * CIN + ci) * 5 + k], acc);
                }
                acc = fmaxf(acc, 0.0f);
                m   = fmaxf(m, acc);
            }
            outb[(co * TPOOL + tp) * TS] = m;
        }
    }
}

__global__ __launch_bounds__(32) void conv_chain_kernel(
    const float* __restrict__ x,                                  // [256,365,399]
    const float* __restrict__ W1, const float* __restrict__ b1,   // [399,3,1,5],[399,3]
    const float* __restrict__ W2, const float* __restrict__ b2,   // [399,3,3,5],[399,3]
    const float* __restrict__ W3, const float* __restrict__ b3,
    const float* __restrict__ W4, const float* __restrict__ b4,
    float* __restrict__ c)                                        // [256,1297]
{
    extern __shared__ float smem[];          // 365*TS + 357*TS floats = 92416 B
    const int lane = threadIdx.x;
    const int pblk = blockIdx.x % NPB;
    const int b    = blockIdx.x / NPB;
    const int p    = pblk * 32 + lane;
    const bool valid = (p < 399);
    const int pc   = valid ? p : 0;

    float* A  = smem + lane;                 // 365 rows, stride TS
    float* Bf = smem + 365 * TS + lane;      // 357 rows, stride TS

    // Stage input: x[b, t, p] — consecutive lanes -> consecutive p -> coalesced
#pragma unroll 1
    for (int t = 0; t < 365; ++t)
        A[t * TS] = valid ? x[(b * 365 + t) * 399 + p] : 0.0f;

    lc_stage<1, 365, 119>(W1 + pc * 15, b1 + pc * 3, A,  Bf);
    lc_stage<3, 119,  37>(W2 + pc * 45, b2 + pc * 3, Bf, A );
    lc_stage<3,  37,  10>(W3 + pc * 45, b3 + pc * 3, A,  Bf);
    lc_stage<3,  10,   1>(W4 + pc * 45, b4 + pc * 3, Bf, A );

    if (valid) {
#pragma unroll
        for (int co = 0; co < 3; ++co)
            c[b * 1297 + co * 399 + p] = A[co * TS];
    }
}

// ---------------------------------------------------------------------------
// f32 GEMM via V_WMMA_F32_16X16X4_F32, branch-free inner loop.
// One wave computes one 16x16 tile of C = A[M,K] * W[K,N] (+bias)(+ReLU).
//  - ragged N handled by clamping the load column to N-1 (stores guarded);
//  - ragged K handled by a single peeled tail quad with clamped addresses
//    and value selects (no exec-mask branches anywhere in the hot loop).
// A-frag (16x4 f32, per ISA layout): lanes 0-15 hold K=k0,k0+1 for row=lane;
// lanes 16-31 hold K=k0+2,k0+3 for row=lane-16.
// C/D frag: VGPR j -> row = tileM*16 + j + 8*(lane>>4), col = tileN*16 + (lane&15).
// grid = (ceil(N/16), M/16), block = 32.
// ---------------------------------------------------------------------------
__global__ __launch_bounds__(32) void gemm_wmma_f32(
    const float* __restrict__ Amat, int lda,   // [M,K], row-major
    const float* __restrict__ Wmat,            // [K,N], row-major (ld = N)
    const float* __restrict__ bias,            // [N] or nullptr
    float* __restrict__ Cmat, int ldc,         // [M,N] (row stride ldc)
    int M, int N, int K, int relu)
{
    const int lane  = threadIdx.x;
    const int half  = lane >> 4;     // 0: K pair {k0,k0+1}; 1: K pair {k0+2,k0+3}
    const int l16   = lane & 15;
    const int tileN = blockIdx.x;
    const int tileM = blockIdx.y;
    const int row   = tileM * 16 + l16;           // A row this lane feeds (M=256, always valid)
    const int col   = tileN * 16 + l16;           // C column this lane owns
    const bool cv   = (col < N);
    const int colL  = cv ? col : (N - 1);         // clamped load column

    const float* Ap = Amat + (size_t)row * lda + 2 * half;  // -> A[row, 2*half]
    const float* Wp = Wmat + (size_t)(2 * half) * N + colL; // -> W[2*half, colL]

    v8f acc = {};
    const int K4 = K & ~3;

#pragma unroll 2
    for (int k0 = 0; k0 < K4; k0 += 4) {
        v2f a, bfr;
        a.x   = Ap[k0];
        a.y   = Ap[k0 + 1];
        bfr.x = Wp[(size_t)k0 * N];
        bfr.y = Wp[(size_t)(k0 + 1) * N];
        acc = __builtin_amdgcn_wmma_f32_16x16x4_f32(
            false, a, false, bfr, (short)0, acc, false, false);
    }

    if (K4 < K) {  // ragged tail quad: clamped addresses + selects, no branches
        const int ka  = K4 + 2 * half;
        const int k0c = (ka     < K) ? ka     : (K - 1);
        const int k1c = (ka + 1 < K) ? ka + 1 : (K - 1);
        v2f a, bfr;
        a.x   = (ka     < K) ? Amat[(size_t)row * lda + k0c] : 0.0f;
        a.y   = (ka + 1 < K) ? Amat[(size_t)row * lda + k1c] : 0.0f;
        bfr.x = (ka     < K) ? Wmat[(size_t)k0c * N + colL]  : 0.0f;
        bfr.y = (ka + 1 < K) ? Wmat[(size_t)k1c * N + colL]  : 0.0f;
        acc = __builtin_amdgcn_wmma_f32_16x16x4_f32(
            false, a, false, bfr, (short)0, acc, false, false);
    }

    const float bv = bias ? bias[colL] : 0.0f;
    if (cv) {
#pragma unroll
        for (int j = 0; j < 8; ++j) {
            const int r = tileM * 16 + j + 8 * half;
            float v = acc[j] + bv;
            if (relu) v = fmaxf(v, 0.0f);
            Cmat[(size_t)r * ldc + col] = v;
        }
    }
}

// ---------------------------------------------------------------------------
// Head: out[b] = g3[b,:20] @ fW + fb
// ---------------------------------------------------------------------------
__global__ __launch_bounds__(256) void final_kernel(
    const float* __restrict__ g3, const float* __restrict__ fW,
    const float* __restrict__ fb, float* __restrict__ out)
{
    const int b = blockIdx.x * blockDim.x + threadIdx.x;
    if (b < 256) {
        float acc = fb[0];
#pragma unroll
        for (int j = 0; j < 20; ++j) acc = fmaf(g3[b * 20 + j], fW[j], acc);
        out[b] = acc;
    }
}

// ---------------------------------------------------------------------------
// Launch
// ---------------------------------------------------------------------------
extern "C" void kernel_launch(void* const* d_in, const int* in_sizes, int n_in,
                              void* d_out, int out_size, void* d_ws, size_t ws_size,
                              hipStream_t stream) {
    (void)in_sizes; (void)n_in; (void)out_size; (void)ws_size;

    const float* x   = (const float*)d_in[0];   // [256,365,399]
    const float* x2  = (const float*)d_in[1];   // [256,1307]
    const float* W1  = (const float*)d_in[2];
    const float* b1  = (const float*)d_in[3];
    const float* W2  = (const float*)d_in[4];
    const float* b2  = (const float*)d_in[5];
    const float* W3  = (const float*)d_in[6];
    const float* b3  = (const float*)d_in[7];
    const float* W4  = (const float*)d_in[8];
    const float* b4  = (const float*)d_in[9];
    const float* lW1 = (const float*)d_in[10];  // [1307,700]
    const float* lb1 = (const float*)d_in[11];
    const float* lW2 = (const float*)d_in[12];  // [700,200]
    const float* lb2 = (const float*)d_in[13];
    const float* lW3 = (const float*)d_in[14];  // [200,100]
    const float* lb3 = (const float*)d_in[15];
    const float* cW1 = (const float*)d_in[16];  // [1297,500]
    const float* cb1 = (const float*)d_in[17];
    const float* cW2 = (const float*)d_in[18];  // [500,100]
    const float* cb2 = (const float*)d_in[19];
    const float* cW3 = (const float*)d_in[20];  // [100,20]
    const float* cb3 = (const float*)d_in[21];
    const float* fW  = (const float*)d_in[22];  // [20,1]
    const float* fb  = (const float*)d_in[23];  // [1]
    float* out = (float*)d_out;                 // [256,1]

    // Workspace carve-up (floats): total ~2.9 MB
    float* c  = (float*)d_ws;        // [256,1297] concat: enc1 (0..1196) | enc2 (1197..1296)
    float* e1 = c  + 256 * 1297;     // [256,700]
    float* e2 = e1 + 256 * 700;      // [256,200]
    float* g1 = e2 + 256 * 200;      // [256,500]
    float* g2 = g1 + 256 * 500;      // [256,100]
    float* g3 = g2 + 256 * 100;      // [256,20]

    // 1) Fused conv chain -> enc1 region of concat buffer
    const int convLds = (365 + 357) * TS * (int)sizeof(float);  // 92416 B
    conv_chain_kernel<<<dim3(256 * NPB), dim3(32), convLds, stream>>>(
        x, W1, b1, W2, b2, W3, b3, W4, b4, c);

    // 2) MLP branch: x2 -> e1 -> e2 -> enc2 (written into concat at col 1197)
    gemm_wmma_f32<<<dim3(44, 16), dim3(32), 0, stream>>>(x2, 1307, lW1, lb1, e1, 700,  256, 700, 1307, 1);
    gemm_wmma_f32<<<dim3(13, 16), dim3(32), 0, stream>>>(e1, 700,  lW2, lb2, e2, 200,  256, 200, 700,  1);
    gemm_wmma_f32<<<dim3(7,  16), dim3(32), 0, stream>>>(e2, 200,  lW3, lb3, c + 1197, 1297, 256, 100, 200, 0);

    // 3) Classifier on concat
    gemm_wmma_f32<<<dim3(32, 16), dim3(32), 0, stream>>>(c,  1297, cW1, cb1, g1, 500,  256, 500, 1297, 1);
    gemm_wmma_f32<<<dim3(7,  16), dim3(32), 0, stream>>>(g1, 500,  cW2, cb2, g2, 100,  256, 100, 500,  1);
    gemm_wmma_f32<<<dim3(2,  16), dim3(32), 0, stream>>>(g2, 100,  cW3, cb3, g3, 20,   256, 20,  100,  1);

    // 4) Head
    final_kernel<<<dim3(1), dim3(256), 0, stream>>>(g3, fW, fb, out);
}